// MMCN_20332375179407
// MI455X (gfx1250) — compile-verified
//
#include <hip/hip_runtime.h>
#include <math.h>

// MI455X / gfx1250 implementation of the ADMM deconvolution network.
//  - PSF correlations via 1024x1024 2-D FFT (Stockham radix-2 in LDS); all
//    FFT state (96MB) is L2-resident on MI455X (192MB L2).
//  - 3x3 convs via implicit-GEMM using fp32 WMMA (V_WMMA_F32_16X16X4_F32),
//    per-tap K=Cin GEMMs, branch-free inner loop:
//       ds_load_b64 (A from LDS halo) + global_load_b64 (B from repacked,
//       zero-padded weights in L2) + v_wmma_f32_16x16x4_f32.
//  - Elementwise ADMM / BN kernels are bandwidth-bound streaming kernels.

typedef __attribute__((ext_vector_type(2))) float v2f;
typedef __attribute__((ext_vector_type(8))) float v8f;

#define NFFT   1024
#define NNC    (1 << 20)           // complex elems per 1024x1024 plane
#define IMH    320
#define IMW    320
#define HWPIX  (IMH * IMW)         // 102400
#define BATCH  4
#define CHANS  3
#define NIMG   (BATCH * CHANS)     // 12
#define IMGN   (NIMG * HWPIX)      // 1228800
#define ADMM_K 5

// ------------------------------------------------------------------ utility
__global__ void mmcn_zero(float* __restrict__ p, int n) {
  int i = blockIdx.x * blockDim.x + threadIdx.x;
  if (i < n) p[i] = 0.f;
}

// ------------------------------------------------------------------ FFT path
__global__ void mmcn_embed_x(const float* __restrict__ x, float2* __restrict__ buf) {
  int idx = blockIdx.x * blockDim.x + threadIdx.x;
  if (idx >= NIMG * NNC) return;
  int img = idx >> 20;
  int r   = (idx >> 10) & 1023;
  int c   = idx & 1023;
  float v = 0.f;
  int rr = r - 159, cc = c - 159;
  if (rr >= 0 && rr < IMH && cc >= 0 && cc < IMW)
    v = x[img * HWPIX + rr * IMW + cc];
  buf[idx] = make_float2(v, 0.f);
}

__global__ void mmcn_embed_psf(const float* __restrict__ k, float2* __restrict__ buf, int flip) {
  int idx = blockIdx.x * blockDim.x + threadIdx.x;
  if (idx >= CHANS * NNC) return;
  int img = idx >> 20;
  int r   = (idx >> 10) & 1023;
  int c   = idx & 1023;
  float v = 0.f;
  if (r < IMH && c < IMW) {
    int rr = flip ? (IMH - 1 - r) : r;
    int cc = flip ? (IMW - 1 - c) : c;
    v = k[img * HWPIX + rr * IMW + cc];
  }
  buf[idx] = make_float2(v, 0.f);
}

// 1024-point radix-2 Stockham FFT along rows (rowsMode=1) or columns (0).
__global__ __launch_bounds__(256) void mmcn_fft1024(float2* __restrict__ data,
                                                    int rowsMode, int dirSign) {
  __shared__ float2 buf[2][NFFT];
  int line = blockIdx.x;
  int img  = line >> 10;
  int lr   = line & 1023;
  size_t base = (size_t)img << 20;
  int stride;
  if (rowsMode) { base += (size_t)lr << 10; stride = 1; }
  else          { base += (size_t)lr;       stride = NFFT; }

  for (int i = threadIdx.x; i < NFFT; i += blockDim.x)
    buf[0][i] = data[base + (size_t)i * stride];
  __syncthreads();

  int src = 0;
  int l = NFFT / 2, m = 1;
  for (int stage = 0; stage < 10; ++stage) {
    int dst = src ^ 1;
    for (int p = threadIdx.x; p < NFFT / 2; p += blockDim.x) {
      int j = p / m;
      int kk = p - j * m;
      float ang = -6.28318530717958648f * (float)dirSign * (float)j / (float)(2 * l);
      float sw, cw;
      __sincosf(ang, &sw, &cw);
      float2 a = buf[src][kk + j * m];
      float2 b = buf[src][kk + j * m + l * m];
      float2 s = make_float2(a.x + b.x, a.y + b.y);
      float2 d = make_float2(a.x - b.x, a.y - b.y);
      buf[dst][kk + 2 * j * m]     = s;
      buf[dst][kk + 2 * j * m + m] = make_float2(cw * d.x - sw * d.y, cw * d.y + sw * d.x);
    }
    __syncthreads();
    src = dst;
    l >>= 1; m <<= 1;
  }
  for (int i = threadIdx.x; i < NFFT; i += blockDim.x)
    data[base + (size_t)i * stride] = buf[src][i];
}

__global__ void mmcn_cmul(float2* __restrict__ buf, const float2* __restrict__ kf) {
  int idx = blockIdx.x * blockDim.x + threadIdx.x;
  if (idx >= NIMG * NNC) return;
  int c = (idx >> 20) % CHANS;
  float2 a = buf[idx];
  float2 b = kf[(c << 20) + (idx & (NNC - 1))];
  buf[idx] = make_float2(a.x * b.x - a.y * b.y, a.x * b.y + a.y * b.x);
}

__global__ void mmcn_crop(const float2* __restrict__ buf, float* __restrict__ out) {
  int idx = blockIdx.x * blockDim.x + threadIdx.x;
  if (idx >= IMGN) return;
  int img = idx / HWPIX;
  int p   = idx - img * HWPIX;
  int h = p / IMW, w = p - h * IMW;
  out[idx] = buf[((size_t)img << 20) + (size_t)(h + 319) * NFFT + (w + 319)].x
             * (1.f / 1048576.f);
}

// ------------------------------------------------------------------ ADMM
__global__ void mmcn_admm_xyz(const float* __restrict__ b, const float* __restrict__ s,
                              const float* __restrict__ Hs,
                              const float* __restrict__ ux, const float* __restrict__ uy,
                              const float* __restrict__ uz,
                              float* __restrict__ x, float* __restrict__ y,
                              float* __restrict__ z,
                              const float* __restrict__ rxp, const float* __restrict__ ryp,
                              const float* __restrict__ rzp, const float* __restrict__ lamp,
                              int kk, int n) {
  int i = blockIdx.x * blockDim.x + threadIdx.x;
  if (i >= n) return;
  float rx = rxp[kk], ry = ryp[kk], rz = rzp[kk], lam = lamp[kk];
  x[i] = (b[i] + rx * Hs[i] - ux[i]) / (1.f + rx);
  float yy = s[i] + uy[i] / ry;
  y[i] = yy > 0.f ? yy : 0.f;
  float zz = s[i] + uz[i] / rz;
  float t = fabsf(zz) - lam / rz;
  t = t > 0.f ? t : 0.f;
  z[i] = zz >= 0.f ? t : -t;
}

__global__ void mmcn_admm_snew(const float* __restrict__ Htx, const float* __restrict__ y,
                               const float* __restrict__ uy, const float* __restrict__ z,
                               const float* __restrict__ uz, float* __restrict__ snew,
                               const float* __restrict__ rxp, const float* __restrict__ ryp,
                               const float* __restrict__ rzp, int kk, int n) {
  int i = blockIdx.x * blockDim.x + threadIdx.x;
  if (i >= n) return;
  float rx = rxp[kk], ry = ryp[kk], rz = rzp[kk];
  snew[i] = (rx * Htx[i] + ry * y[i] - uy[i] + rz * z[i] - uz[i]) / (rx + ry + rz + 1e-6f);
}

__global__ void mmcn_admm_dual(const float* __restrict__ Hs2, const float* __restrict__ x,
                               const float* __restrict__ y, const float* __restrict__ z,
                               const float* __restrict__ snew,
                               float* __restrict__ ux, float* __restrict__ uy,
                               float* __restrict__ uz,
                               const float* __restrict__ rxp, const float* __restrict__ ryp,
                               const float* __restrict__ rzp, int kk, int n) {
  int i = blockIdx.x * blockDim.x + threadIdx.x;
  if (i >= n) return;
  float rx = rxp[kk], ry = ryp[kk], rz = rzp[kk];
  ux[i] += rx * (Hs2[i] - x[i]);
  uy[i] += ry * (snew[i] - y[i]);
  uz[i] += rz * (snew[i] - z[i]);
}

// ------------------------------------------------------------------ weight repack
// wP[tap][pair][co16][2]: zero-padded to Cin4 = ceil4(Cin), Cout16 = ceil16(Cout).
// Element [tap][p][co][j] = w[co][2p+j][tap/3][tap%3]  (OIHW flat (co*Cin+ci)*9+tap).
__global__ void mmcn_repack_w(const float* __restrict__ w, float* __restrict__ wp,
                              int Cin, int Cout) {
  const int Cin4   = (Cin + 3) & ~3;
  const int Cout16 = (Cout + 15) & ~15;
  const int halfPairs = Cin4 >> 1;
  const int total = 9 * halfPairs * Cout16 * 2;
  int idx = blockIdx.x * blockDim.x + threadIdx.x;
  if (idx >= total) return;
  int j  = idx & 1;
  int t  = idx >> 1;
  int co = t % Cout16;  t /= Cout16;
  int p  = t % halfPairs;
  int tap = t / halfPairs;
  int ci = 2 * p + j;
  float v = 0.f;
  if (ci < Cin && co < Cout)
    v = w[((size_t)co * Cin + ci) * 9 + tap];
  wp[idx] = v;
}

// ------------------------------------------------------------------ conv3x3 (fp32 WMMA)
// grid = (W/16, H, B); block = 32 * (Cout16/16) threads (<=256).
// Per-tap GEMM: K = Cin4, A[m][ci] from LDS halo, B[ci][co] from repacked
// weights. fp32 WMMA 16x16x4: lanes 0-15 hold K = k0,k0+1; lanes 16-31 K+2,+3.
__global__ __launch_bounds__(256) void mmcn_conv3x3_wmma(
    const float* __restrict__ x,    // [B, Cin, H, W]
    const float* __restrict__ wp,   // repacked weights (see mmcn_repack_w)
    const float* __restrict__ bias, // [Cout]
    float* __restrict__ out,        // [B, ldC, H, W], channels written at coOff
    int Cin, int Cout, int ldC, int coOff) {
  const int Cin4   = (Cin + 3) & ~3;
  const int Cout16 = (Cout + 15) & ~15;
  const int LDSP   = Cin4 + 2;            // even stride: b64-aligned, conflict-mitigating
  __shared__ float patch[3 * 18 * (144 + 2)];   // max 30.8KB of 320KB/WGP
  const int wt = blockIdx.x * 16;
  const int h  = blockIdx.y;
  const int n  = blockIdx.z;
  const int tid = threadIdx.x;

  // Stage halo transposed: patch[(dy*18+xx)*LDSP + ci]; pad channels zeroed.
  for (int i = tid; i < Cin4 * 54; i += blockDim.x) {
    int ci = i / 54;
    int r  = i - ci * 54;
    int dy = r / 18;
    int xx = r - dy * 18;
    int gy = h + dy - 1;
    int gx = wt - 1 + xx;
    float v = 0.f;
    if (ci < Cin && gy >= 0 && gy < IMH && gx >= 0 && gx < IMW)
      v = x[(((size_t)n * Cin + ci) * IMH + gy) * IMW + gx];
    patch[(dy * 18 + xx) * LDSP + ci] = v;
  }
  __syncthreads();

  const int wv   = tid >> 5;
  const int lane = tid & 31;
  const int m  = lane & 15;               // A row M (w offset) / B col N (cout)
  const int hi = lane >> 4;               // K half select
  const int co = wv * 16 + m;
  const int halfPairs = Cin4 >> 1;
  const size_t bStep = (size_t)2 * Cout16 * 2;   // 4 channels worth of pairs

  v8f acc = {};
  for (int tap = 0; tap < 9; ++tap) {
    int dy = tap / 3, dx = tap - dy * 3;
    const float* aptr = &patch[(dy * 18 + m + dx) * LDSP + 2 * hi];
    const float* bptr = wp + (((size_t)tap * halfPairs + hi) * Cout16 + co) * 2;
    __builtin_prefetch(bptr + bStep, 0, 3);       // global_prefetch_b8 (L2 stream)
    #pragma unroll 2
    for (int c0 = 0; c0 < Cin4; c0 += 4) {
      v2f a = *(const v2f*)aptr;                  // ds_load_b64
      v2f b = *(const v2f*)bptr;                  // global_load_b64 (coalesced)
      acc = __builtin_amdgcn_wmma_f32_16x16x4_f32(false, a, false, b,
                                                  (short)0, acc, false, false);
      aptr += 4;
      bptr += bStep;
    }
  }

  if (co >= Cout) return;
  float bv = bias[co];
  // D layout: lane (hi,m): VGPR r -> element (M = hi*8 + r, N = m).
  size_t obase = (((size_t)n * ldC + (coOff + co)) * IMH + h) * IMW + wt + hi * 8;
  #pragma unroll
  for (int r = 0; r < 8; ++r)
    out[obase + r] = acc[r] + bv;
}

// ------------------------------------------------------------------ batch norm
__global__ __launch_bounds__(256) void mmcn_bnstats(const float* __restrict__ buf,
                                                    int ldC, int coOff,
                                                    float* __restrict__ stats) {
  int c = blockIdx.y;
  __shared__ float ss[256], sq[256];
  float s = 0.f, q = 0.f;
  const int per = BATCH * HWPIX;
  for (int i = blockIdx.x * blockDim.x + threadIdx.x; i < per; i += gridDim.x * blockDim.x) {
    int n = i / HWPIX;
    int p = i - n * HWPIX;
    float v = buf[((size_t)n * ldC + coOff + c) * HWPIX + p];
    s += v; q += v * v;
  }
  ss[threadIdx.x] = s; sq[threadIdx.x] = q;
  __syncthreads();
  for (int o = 128; o > 0; o >>= 1) {
    if ((int)threadIdx.x < o) {
      ss[threadIdx.x] += ss[threadIdx.x + o];
      sq[threadIdx.x] += sq[threadIdx.x + o];
    }
    __syncthreads();
  }
  if (threadIdx.x == 0) {
    atomicAdd(&stats[2 * c],     ss[0]);
    atomicAdd(&stats[2 * c + 1], sq[0]);
  }
}

__global__ void mmcn_bnapply(float* __restrict__ buf, int ldC, int coOff, int C,
                             const float* __restrict__ g, const float* __restrict__ be,
                             const float* __restrict__ stats) {
  int idx = blockIdx.x * blockDim.x + threadIdx.x;
  int total = BATCH * C * HWPIX;
  if (idx >= total) return;
  int n   = idx / (C * HWPIX);
  int rem = idx - n * C * HWPIX;
  int c   = rem / HWPIX;
  int p   = rem - c * HWPIX;
  const float cnt = (float)(BATCH * HWPIX);
  float mu  = stats[2 * c] / cnt;
  float var = stats[2 * c + 1] / cnt - mu * mu;
  float inv = rsqrtf(var + 1e-5f);
  size_t o = ((size_t)n * ldC + coOff + c) * HWPIX + p;
  float v = g[c] * (buf[o] - mu) * inv + be[c];
  buf[o] = v > 0.f ? v : 0.f;
}

// ------------------------------------------------------------------ final 1x1 conv + clip
__global__ void mmcn_fin(const float* __restrict__ x, const float* __restrict__ w,
                         const float* __restrict__ b, float* __restrict__ out) {
  int idx = blockIdx.x * blockDim.x + threadIdx.x;
  const int total = BATCH * CHANS * HWPIX;
  if (idx >= total) return;
  int n   = idx / (CHANS * HWPIX);
  int rem = idx - n * CHANS * HWPIX;
  int co  = rem / HWPIX;
  int p   = rem - co * HWPIX;
  const float* xp = x + (size_t)n * 32 * HWPIX + p;
  float acc = b[co];
  #pragma unroll 8
  for (int ci = 0; ci < 32; ++ci)
    acc += xp[(size_t)ci * HWPIX] * w[co * 32 + ci];
  out[idx] = fminf(fmaxf(acc, 0.f), 1.f);
}

// ------------------------------------------------------------------ driver
extern "C" void kernel_launch(void* const* d_in, const int* in_sizes, int n_in,
                              void* d_out, int out_size, void* d_ws, size_t ws_size,
                              hipStream_t stream) {
  const float* bimg    = (const float*)d_in[0];
  const float* psf     = (const float*)d_in[1];
  const float* rho_x   = (const float*)d_in[2];
  const float* rho_y   = (const float*)d_in[3];
  const float* rho_z   = (const float*)d_in[4];
  const float* lambd   = (const float*)d_in[5];
  const float* meas_w  = (const float*)d_in[6];
  const float* meas_b  = (const float*)d_in[7];
  const float* meas_g  = (const float*)d_in[8];
  const float* meas_be = (const float*)d_in[9];
  const float* comp_w  = (const float*)d_in[10];
  const float* comp_b  = (const float*)d_in[11];
  const float* comp_g  = (const float*)d_in[12];
  const float* comp_be = (const float*)d_in[13];
  const float* d1_w = (const float*)d_in[14]; const float* d1_b = (const float*)d_in[15];
  const float* d1_g = (const float*)d_in[16]; const float* d1_be = (const float*)d_in[17];
  const float* d2_w = (const float*)d_in[18]; const float* d2_b = (const float*)d_in[19];
  const float* d2_g = (const float*)d_in[20]; const float* d2_be = (const float*)d_in[21];
  const float* d3_w = (const float*)d_in[22]; const float* d3_b = (const float*)d_in[23];
  const float* d3_g = (const float*)d_in[24]; const float* d3_be = (const float*)d_in[25];
  const float* fin_w = (const float*)d_in[26]; const float* fin_b = (const float*)d_in[27];
  float* outp = (float*)d_out;

  // ---- workspace bump allocator (all sizes multiples of 256B)
  char* ws = (char*)d_ws;
  size_t off = 0;
  auto alloc = [&](size_t bytes) -> void* {
    void* p = ws + off;
    off = (off + bytes + 255) & ~(size_t)255;
    return p;
  };
  float2* fbuf  = (float2*)alloc((size_t)NIMG * NNC * sizeof(float2));   //  96 MB
  float2* kfC   = (float2*)alloc((size_t)CHANS * NNC * sizeof(float2));  //  24 MB  FFT(flip psf)
  float2* kfV   = (float2*)alloc((size_t)CHANS * NNC * sizeof(float2));  //  24 MB  FFT(psf)
  float*  suB   = (float*)alloc((size_t)4 * IMGN * sizeof(float));       // s0,ux,uy,uz
  float*  sList = (float*)alloc((size_t)ADMM_K * IMGN * sizeof(float));
  float*  xb    = (float*)alloc((size_t)IMGN * sizeof(float));
  float*  yb    = (float*)alloc((size_t)IMGN * sizeof(float));
  float*  zb    = (float*)alloc((size_t)IMGN * sizeof(float));
  float*  hbuf  = (float*)alloc((size_t)IMGN * sizeof(float));
  float*  feats = (float*)alloc((size_t)BATCH * 144 * HWPIX * sizeof(float)); // 225 MB
  float*  d1o   = (float*)alloc((size_t)BATCH * 128 * HWPIX * sizeof(float)); // 200 MB
  float*  stats = (float*)alloc((size_t)2 * 144 * sizeof(float));
  float*  wpBuf = (float*)alloc((size_t)9 * 144 * 128 * sizeof(float));  // 663KB repack
  // d2/d3 outputs alias the retired FFT scratch (fbuf..suB, 170MB contiguous):
  float* d2o = (float*)d_ws;                                  // 100 MB
  float* d3o = (float*)((char*)d_ws + (size_t)BATCH * 64 * HWPIX * sizeof(float)); // 50 MB

  float* s0 = suB;
  float* ux = suB + (size_t)1 * IMGN;
  float* uy = suB + (size_t)2 * IMGN;
  float* uz = suB + (size_t)3 * IMGN;

  const int TPB = 256;
  auto cdiv = [](int a, int b) { return (a + b - 1) / b; };

  auto fft2 = [&](float2* buf, int nimg, int dirSign) {
    mmcn_fft1024<<<nimg * NFFT, TPB, 0, stream>>>(buf, 1, dirSign);
    mmcn_fft1024<<<nimg * NFFT, TPB, 0, stream>>>(buf, 0, dirSign);
  };
  auto psf_corr = [&](const float* in, float* res, const float2* kf) {
    mmcn_embed_x<<<cdiv(NIMG * NNC, TPB), TPB, 0, stream>>>(in, fbuf);
    fft2(fbuf, NIMG, +1);
    mmcn_cmul<<<cdiv(NIMG * NNC, TPB), TPB, 0, stream>>>(fbuf, kf);
    fft2(fbuf, NIMG, -1);
    mmcn_crop<<<cdiv(IMGN, TPB), TPB, 0, stream>>>(fbuf, res);
  };
  auto conv_bn = [&](const float* xin, int Cin, const float* wgt, const float* bias,
                     float* obuf, int ldC, int coOff, int Cout,
                     const float* g, const float* be) {
    int Cin4 = (Cin + 3) & ~3;
    int Cout16 = (Cout + 15) & ~15;
    int wpN = 9 * Cin4 * Cout16;
    mmcn_repack_w<<<cdiv(wpN, TPB), TPB, 0, stream>>>(wgt, wpBuf, Cin, Cout);
    mmcn_conv3x3_wmma<<<dim3(IMW / 16, IMH, BATCH), 32 * (Cout16 / 16), 0, stream>>>(
        xin, wpBuf, bias, obuf, Cin, Cout, ldC, coOff);
    mmcn_zero<<<1, TPB, 0, stream>>>(stats, 2 * Cout);
    mmcn_bnstats<<<dim3(32, Cout), TPB, 0, stream>>>(obuf, ldC, coOff, stats);
    mmcn_bnapply<<<cdiv(BATCH * Cout * HWPIX, TPB), TPB, 0, stream>>>(
        obuf, ldC, coOff, Cout, g, be, stats);
  };

  // ---- PSF spectra (psf is (3,1,320,320) == (3,320,320) contiguous)
  mmcn_embed_psf<<<cdiv(CHANS * NNC, TPB), TPB, 0, stream>>>(psf, kfC, 1); // corr kernel
  fft2(kfC, CHANS, +1);
  mmcn_embed_psf<<<cdiv(CHANS * NNC, TPB), TPB, 0, stream>>>(psf, kfV, 0); // conv kernel
  fft2(kfV, CHANS, +1);

  // ---- ADMM unroll
  mmcn_zero<<<cdiv(4 * IMGN, TPB), TPB, 0, stream>>>(suB, 4 * IMGN);  // s0,u*=0
  const float* sprev = s0;
  for (int kk = 0; kk < ADMM_K; ++kk) {
    float* snew = sList + (size_t)kk * IMGN;
    psf_corr(sprev, hbuf, kfC);                                       // Hs
    mmcn_admm_xyz<<<cdiv(IMGN, TPB), TPB, 0, stream>>>(
        bimg, sprev, hbuf, ux, uy, uz, xb, yb, zb,
        rho_x, rho_y, rho_z, lambd, kk, IMGN);
    psf_corr(xb, hbuf, kfV);                                          // Ht_x
    mmcn_admm_snew<<<cdiv(IMGN, TPB), TPB, 0, stream>>>(
        hbuf, yb, uy, zb, uz, snew, rho_x, rho_y, rho_z, kk, IMGN);
    psf_corr(snew, hbuf, kfC);                                        // Hs2
    mmcn_admm_dual<<<cdiv(IMGN, TPB), TPB, 0, stream>>>(
        hbuf, xb, yb, zb, snew, ux, uy, uz, rho_x, rho_y, rho_z, kk, IMGN);
    sprev = snew;
  }

  // ---- feature extraction + decoder (WMMA convs)
  conv_bn(bimg, 3, meas_w, meas_b, feats, 144, 0, 24, meas_g, meas_be);
  for (int kk = 0; kk < ADMM_K; ++kk) {
    conv_bn(sList + (size_t)kk * IMGN, 3,
            comp_w + (size_t)kk * 24 * 27, comp_b + (size_t)kk * 24,
            feats, 144, 24 + 24 * kk, 24,
            comp_g + (size_t)kk * 24, comp_be + (size_t)kk * 24);
  }
  conv_bn(feats, 144, d1_w, d1_b, d1o, 128, 0, 128, d1_g, d1_be);
  conv_bn(d1o,   128, d2_w, d2_b, d2o,  64, 0,  64, d2_g, d2_be);
  conv_bn(d2o,    64, d3_w, d3_b, d3o,  32, 0,  32, d3_g, d3_be);
  mmcn_fin<<<cdiv(BATCH * CHANS * HWPIX, TPB), TPB, 0, stream>>>(d3o, fin_w, fin_b, outp);
}